// Summary_net_1dps_det_4638564680191
// MI455X (gfx1250) — compile-verified
//
#include <hip/hip_runtime.h>
#include <math.h>

#define BATCH 131072
#define HID   256
#define WAVES_PER_BLOCK 2

#define XSTR 88    // row stride (floats) for x / a buffers (max c=80, padded)
#define HSTR 264   // row stride for hidden buffer (256, padded)
// per wave: xbuf(16*XSTR) + hbuf(16*HSTR) + abuf(16*XSTR) + scbuf(88)
#define WAVE_F (16*XSTR + 16*HSTR + 16*XSTR + 88)

typedef __attribute__((ext_vector_type(2))) float v2f;
typedef __attribute__((ext_vector_type(8))) float v8f;

struct BlockP {
  const float* W1; const float* W2; const float* b1; const float* b2;
  const float* bn_beta; const float* bn_gamma; const float* bn_mean; const float* bn_var;
  const float* goff; const float* gscale; const float* wperm;
};

struct KArgs {
  const float* x;
  float* out;     // [BATCH*6] x_out, then [BATCH] loss
  BlockP bp[6];
};

// Intra-wave LDS ordering: all producers/consumers are in the same wave32,
// DS ops are in-order per wave; force completion + block compiler motion.
__device__ __forceinline__ void wave_lds_fence() {
  asm volatile("s_wait_dscnt 0" ::: "memory");
}

// A fragment (16x4, f32) from an LDS row-major tile.
// ISA layout: lanes 0-15 hold (M=lane, K=k0+0/1); lanes 16-31 (M=lane-16, K=k0+2/3).
__device__ __forceinline__ v2f frag_a_lds(const float* buf, int stride, int k0,
                                          int kmax, int lane) {
  int m  = lane & 15;
  int kb = k0 + ((lane >> 4) << 1);
  const float* p = buf + m * stride + kb;   // padded stride -> always in-bounds
  float v0 = p[0];
  float v1 = p[1];
  v2f a;
  a.x = (kb     < kmax) ? v0 : 0.0f;
  a.y = (kb + 1 < kmax) ? v1 : 0.0f;
  return a;
}

// B fragment (4x16, f32) from global row-major W[K][N] (ld = row stride).
__device__ __forceinline__ v2f frag_b_glob(const float* W, int ld, int k0, int kmax,
                                           int n0, int ncols, int lane) {
  int n  = n0 + (lane & 15);
  int nn = (n < ncols) ? n : 0;
  int kb = k0 + ((lane >> 4) << 1);
  int k1 = (kb     < kmax) ? kb     : 0;
  int k2 = (kb + 1 < kmax) ? kb + 1 : 0;
  float v0 = W[(size_t)k1 * ld + nn];
  float v1 = W[(size_t)k2 * ld + nn];
  v2f b;
  b.x = ((kb     < kmax) && (n < ncols)) ? v0 : 0.0f;
  b.y = ((kb + 1 < kmax) && (n < ncols)) ? v1 : 0.0f;
  return b;
}

// B fragment where B = M^T, i.e. B[k][n] = M[n*ld + k] (for xo @ w_perm.T).
__device__ __forceinline__ v2f frag_bt_glob(const float* M, int ld, int k0, int kmax,
                                            int n0, int ncols, int lane) {
  int n  = n0 + (lane & 15);
  int nn = (n < ncols) ? n : 0;
  int kb = k0 + ((lane >> 4) << 1);
  int k1 = (kb     < kmax) ? kb     : 0;
  int k2 = (kb + 1 < kmax) ? kb + 1 : 0;
  float v0 = M[(size_t)nn * ld + k1];
  float v1 = M[(size_t)nn * ld + k2];
  v2f b;
  b.x = ((kb     < kmax) && (n < ncols)) ? v0 : 0.0f;
  b.y = ((kb + 1 < kmax) && (n < ncols)) ? v1 : 0.0f;
  return b;
}

__device__ __forceinline__ v8f wmma4(v2f a, v2f b, v8f c) {
  return __builtin_amdgcn_wmma_f32_16x16x4_f32(
      /*neg_a=*/false, a, /*neg_b=*/false, b,
      /*c_mod=*/(short)0, c, /*reuse_a=*/false, /*reuse_b=*/false);
}

// One coupling block on a 16-row tile held in xbuf. Returns per-row log-det
// (row = lane & 15; valid in all lanes). xbuf in/out has C columns.
template <int C>
__device__ float run_block(const BlockP& p, float* xbuf, float* hbuf, float* abuf,
                           float* scbuf, int lane) {
  constexpr int S2  = C >> 1;
  constexpr int S1  = C - S2;
  constexpr int OC  = 2 * S2;
  constexpr int NT2 = (OC + 15) >> 4;   // GEMM2 column tiles (<=5)
  constexpr int NTP = (C + 15) >> 4;    // perm GEMM column tiles (<=5)
  const int n16 = lane & 15;
  const int mb  = (lane >> 4) << 3;     // C/D frag: lanes<16 rows 0..7, else 8..15
  const int r   = lane & 15;

  // ---- per-wave scale precompute: sc[ch] = 0.2*softplus(0.5*gs), lsum = sum log ----
  float lpart = 0.0f;
  for (int ch = lane; ch < C; ch += 32) {
    float sc = 0.2f * log1pf(expf(0.5f * p.gscale[ch]));
    scbuf[ch] = sc;
    lpart += logf(sc);
  }
  float lsum = lpart;
#pragma unroll
  for (int off = 16; off > 0; off >>= 1) lsum += __shfl_xor(lsum, off, 32);

  // ---- GEMM1: H = gelu(x1 @ W1 + b1); x1 = xbuf[:, 0:S1], W1 (S1 x 256) ----
  // Multi-accumulator: 4 column tiles per A-fragment load.
  for (int g = 0; g < HID / 64; ++g) {
    v8f acc[4];
#pragma unroll
    for (int j = 0; j < 4; ++j) { v8f z = {}; acc[j] = z; }
    for (int k0 = 0; k0 < S1; k0 += 4) {
      v2f a = frag_a_lds(xbuf, XSTR, k0, S1, lane);
#pragma unroll
      for (int j = 0; j < 4; ++j) {
        v2f b = frag_b_glob(p.W1, HID, k0, S1, (g * 4 + j) * 16, HID, lane);
        acc[j] = wmma4(a, b, acc[j]);
      }
    }
#pragma unroll
    for (int j = 0; j < 4; ++j) {
      int n = (g * 4 + j) * 16 + n16;
      float bias = p.b1[n];
#pragma unroll
      for (int v = 0; v < 8; ++v) {
        float h = acc[j][v] + bias;
        h = 0.5f * h * (1.0f + erff(h * 0.70710678118654752f));  // exact gelu
        hbuf[(mb + v) * HSTR + n] = h;
      }
    }
  }
  wave_lds_fence();

  // ---- GEMM2: a = 0.1 * BN(H @ W2 + b2); out cols = OC ----
  // One A-fragment feeds all NT2 column tiles per k-step.
  {
    v8f acc[NT2];
#pragma unroll
    for (int j = 0; j < NT2; ++j) { v8f z = {}; acc[j] = z; }
    for (int k0 = 0; k0 < HID; k0 += 4) {
      v2f a = frag_a_lds(hbuf, HSTR, k0, HID, lane);
#pragma unroll
      for (int j = 0; j < NT2; ++j) {
        v2f b = frag_b_glob(p.W2, OC, k0, HID, j * 16, OC, lane);
        acc[j] = wmma4(a, b, acc[j]);
      }
    }
#pragma unroll
    for (int j = 0; j < NT2; ++j) {
      int n = j * 16 + n16;
      if (n < OC) {
        float bias = p.b2[n];
        float g    = p.bn_gamma[n] * rsqrtf(p.bn_var[n] + 1e-5f);
        float mean = p.bn_mean[n];
        float beta = p.bn_beta[n];
#pragma unroll
        for (int v = 0; v < 8; ++v) {
          float h = acc[j][v] + bias;
          abuf[(mb + v) * XSTR + n] = 0.1f * ((h - mean) * g + beta);
        }
      }
    }
  }
  wave_lds_fence();

  // ---- coupling: 2 lanes per row (lane r: cols [0,half), lane r+16: [half,S2)) ----
  constexpr int HALF = (S2 + 1) >> 1;
  float jpart = 0.0f;
  {
    const float* xr2 = xbuf + r * XSTR;
    const float* ar  = abuf + r * XSTR;
    float*       xo  = hbuf + r * HSTR;
    const int i0 = (lane < 16) ? 0 : HALF;
    const int i1 = (lane < 16) ? HALF : S2;
    for (int i = i0; i < i1; ++i) {
      float sj = 2.0f * tanhf(ar[i]);            // CLAMP * tanh(a[:, :S2])
      jpart += sj;
      xo[S1 + i] = xr2[S1 + i] * expf(sj) + ar[S2 + i];
    }
  }
  float jrow = jpart + __shfl_xor(jpart, 16, 32);  // per-row total in both lanes
  wave_lds_fence();

  // ---- xo = concat(x1, x2') * scale + offset, all 32 lanes strided ----
  for (int t = lane; t < 16 * C; t += 32) {
    int rr = t / C, ch = t % C;
    float val = (ch < S1) ? xbuf[rr * XSTR + ch] : hbuf[rr * HSTR + ch];
    hbuf[rr * HSTR + ch] = val * scbuf[ch] + p.goff[ch];
  }
  wave_lds_fence();

  // ---- perm GEMM: xbuf = xo @ wperm^T (C x C) ----
  {
    v8f acc[NTP];
#pragma unroll
    for (int j = 0; j < NTP; ++j) { v8f z = {}; acc[j] = z; }
    for (int k0 = 0; k0 < C; k0 += 4) {
      v2f a = frag_a_lds(hbuf, HSTR, k0, C, lane);
#pragma unroll
      for (int j = 0; j < NTP; ++j) {
        v2f b = frag_bt_glob(p.wperm, C, k0, C, j * 16, C, lane);
        acc[j] = wmma4(a, b, acc[j]);
      }
    }
#pragma unroll
    for (int j = 0; j < NTP; ++j) {
      int n = j * 16 + n16;
      if (n < C) {
#pragma unroll
        for (int v = 0; v < 8; ++v) xbuf[(mb + v) * XSTR + n] = acc[j][v];
      }
    }
  }
  wave_lds_fence();
  return jrow + lsum;
}

__global__ void __launch_bounds__(32 * WAVES_PER_BLOCK)
flow_kernel(KArgs args) {
  extern __shared__ float smem[];
  const int lane = threadIdx.x & 31;
  const int wid  = threadIdx.x >> 5;
  const int tile = blockIdx.x * WAVES_PER_BLOCK + wid;   // 8192 tiles exactly

  float* xbuf  = smem + wid * WAVE_F;
  float* hbuf  = xbuf + 16 * XSTR;
  float* abuf  = hbuf + 16 * HSTR;
  float* scbuf = abuf + 16 * XSTR;

  // Load 16 rows x 80 features (coalesced within the wave).
  {
    const float* src = args.x + (size_t)tile * 16 * 80;
    for (int t = lane; t < 16 * 80; t += 32) {
      xbuf[(t / 80) * XSTR + (t % 80)] = src[t];
    }
  }
  wave_lds_fence();

  float loss = 0.0f;   // valid in lanes 0..15 (one row each)
  float det;
  float* xr = xbuf + (lane & 15) * XSTR;

  // Block 0 (c=80), then loss + downsample -> 40
  det = run_block<80>(args.bp[0], xbuf, hbuf, abuf, scbuf, lane);
  if (lane < 16) {
    float ss = 0.0f;
    for (int ch = 0; ch < 80; ++ch) ss += xr[ch] * xr[ch];
    loss += 0.5f * ss - det;
    for (int i = 0; i < 40; ++i) xr[i] = 0.5f * (xr[2 * i] + xr[2 * i + 1]);
  }
  wave_lds_fence();

  // Block 1 (c=40), loss + down -> 20
  det = run_block<40>(args.bp[1], xbuf, hbuf, abuf, scbuf, lane);
  if (lane < 16) {
    float ss = 0.0f;
    for (int ch = 0; ch < 40; ++ch) ss += xr[ch] * xr[ch];
    loss += 0.5f * ss - det;
    for (int i = 0; i < 20; ++i) xr[i] = 0.5f * (xr[2 * i] + xr[2 * i + 1]);
  }
  wave_lds_fence();

  // Block 2 (c=20), loss + down -> 10
  det = run_block<20>(args.bp[2], xbuf, hbuf, abuf, scbuf, lane);
  if (lane < 16) {
    float ss = 0.0f;
    for (int ch = 0; ch < 20; ++ch) ss += xr[ch] * xr[ch];
    loss += 0.5f * ss - det;
    for (int i = 0; i < 10; ++i) xr[i] = 0.5f * (xr[2 * i] + xr[2 * i + 1]);
  }
  wave_lds_fence();

  // Block 3 (c=10), loss + affine trick -> 8
  det = run_block<10>(args.bp[3], xbuf, hbuf, abuf, scbuf, lane);
  if (lane < 16) {
    float ss = 0.0f;
    for (int ch = 0; ch < 10; ++ch) ss += xr[ch] * xr[ch];
    loss += 0.5f * ss - det;
    float tm = xr[9], ta = xr[8];
    for (int i = 0; i < 8; ++i) xr[i] = xr[i] * tm + ta;
  }
  wave_lds_fence();

  // Block 4 (c=8), loss + affine trick -> 6
  det = run_block<8>(args.bp[4], xbuf, hbuf, abuf, scbuf, lane);
  if (lane < 16) {
    float ss = 0.0f;
    for (int ch = 0; ch < 8; ++ch) ss += xr[ch] * xr[ch];
    loss += 0.5f * ss - det;
    float tm = xr[7], ta = xr[6];
    for (int i = 0; i < 6; ++i) xr[i] = xr[i] * tm + ta;
  }
  wave_lds_fence();

  // Block 5 (c=6): sigmoid, then loss; write outputs.
  det = run_block<6>(args.bp[5], xbuf, hbuf, abuf, scbuf, lane);
  if (lane < 16) {
    const int row = tile * 16 + lane;
    float ss = 0.0f;
    for (int ch = 0; ch < 6; ++ch) {
      float v = 1.0f / (1.0f + expf(-xr[ch]));
      args.out[(size_t)row * 6 + ch] = v;
      ss += v * v;
    }
    loss += 0.5f * ss - det;
    args.out[(size_t)BATCH * 6 + row] = loss;
  }
}

extern "C" void kernel_launch(void* const* d_in, const int* in_sizes, int n_in,
                              void* d_out, int out_size, void* d_ws, size_t ws_size,
                              hipStream_t stream) {
  (void)in_sizes; (void)n_in; (void)out_size; (void)d_ws; (void)ws_size;
  KArgs a;
  a.x   = (const float*)d_in[0];
  a.out = (float*)d_out;
  // JAX pytree flattening sorts dict keys: per block ->
  // W1, W2, b1, b2, bn_beta, bn_gamma, bn_mean, bn_var,
  // global_offset, global_scale, w_perm
  for (int i = 0; i < 6; ++i) {
    const float* const* p = (const float* const*)(d_in + 1 + i * 11);
    a.bp[i].W1 = p[0];       a.bp[i].W2 = p[1];
    a.bp[i].b1 = p[2];       a.bp[i].b2 = p[3];
    a.bp[i].bn_beta = p[4];  a.bp[i].bn_gamma = p[5];
    a.bp[i].bn_mean = p[6];  a.bp[i].bn_var = p[7];
    a.bp[i].goff = p[8];     a.bp[i].gscale = p[9];
    a.bp[i].wperm = p[10];
  }
  const int grid = BATCH / (16 * WAVES_PER_BLOCK);     // 4096
  const int blk  = 32 * WAVES_PER_BLOCK;               // 64
  const size_t shmem = (size_t)WAVE_F * WAVES_PER_BLOCK * sizeof(float);  // ~57 KB
  hipLaunchKernelGGL(flow_kernel, dim3(grid), dim3(blk), shmem, stream, a);
}